// QuantumFusionLayer_67233418051867
// MI455X (gfx1250) — compile-verified
//
#include <hip/hip_runtime.h>

typedef float v2f __attribute__((ext_vector_type(2)));
typedef float v8f __attribute__((ext_vector_type(8)));

#define ROWS 16      // batch rows per workgroup (= WMMA M)
#define RS   2052    // padded LDS row stride in floats (2048 + 4 pad); 8208 B, 16B-aligned

__device__ __forceinline__ float hsign(int a, int b) {
    // Unnormalized Hadamard entry: (-1)^popcount(a & b)
    return (__popc(a & b) & 1) ? -1.0f : 1.0f;
}

__global__ __launch_bounds__(256)
void qwht_fusion_kernel(const float* __restrict__ x1,
                        const float* __restrict__ x2,
                        float* __restrict__ out) {
    extern __shared__ float lds[];
    const int tid  = threadIdx.x;
    const int lane = tid & 31;
    const int wave = tid >> 5;
    const int rowsBase = blockIdx.x * ROWS;

    // ---- Async global -> LDS staging (no VGPR round-trip, ASYNCcnt path) ----
    // Iteration parity selects x1/x2 uniformly; each lane moves one float4.
    #pragma unroll 1
    for (int it = 0; it < 32; ++it) {
        const int r    = it >> 1;
        const int half = it & 1;
        const float4* g = (const float4*)(half ? x2 : x1)
                          + (size_t)(rowsBase + r) * 256 + tid;
        const unsigned ldsOff =
            (unsigned)((r * RS + (half * 256 + tid) * 4) * sizeof(float));
        asm volatile("global_load_async_to_lds_b128 %0, %1, off"
                     :: "v"(ldsOff), "v"(g) : "memory");
    }
    asm volatile("s_wait_asynccnt 0" ::: "memory");
    __syncthreads();

    // WMMA fragment lane mapping (16x16x4 f32):
    //   A: lanes 0-15 hold M=lane, K={4c+0,4c+1}; lanes 16-31 M=lane-16, K={4c+2,4c+3}
    //   B: VGPR0 row 4c+koff (N=lane&15), VGPR1 row 4c+koff+1
    //   C/D: VGPR v -> row (v + 8*(lane>=16)), col lane&15
    const int r    = lane & 15;
    const int hi   = lane >> 4;
    const int koff = hi * 2;
    const int n    = lane & 15;

    // ---- Stage 1: H16 on state bits [3:0] ----
    #pragma unroll 1
    for (int it = 0; it < 16; ++it) {
        const int base = (it * 8 + wave) * 16;
        v8f acc = {};
        #pragma unroll
        for (int ch = 0; ch < 4; ++ch) {
            int k0 = ch * 4 + koff;
            v2f a, b;
            a.x = lds[r * RS + base + k0];
            a.y = lds[r * RS + base + k0 + 1];
            b.x = hsign(k0, n);
            b.y = hsign(k0 + 1, n);
            acc = __builtin_amdgcn_wmma_f32_16x16x4_f32(false, a, false, b,
                                                        (short)0, acc, false, false);
        }
        #pragma unroll
        for (int v = 0; v < 8; ++v)
            lds[(v + hi * 8) * RS + base + n] = acc[v];
    }
    __syncthreads();

    // ---- Stage 2: H16 on state bits [7:4] (stride-16 columns) ----
    #pragma unroll 1
    for (int it = 0; it < 16; ++it) {
        const int t = it * 8 + wave;
        const int base = (t >> 4) * 256 + (t & 15);
        v8f acc = {};
        #pragma unroll
        for (int ch = 0; ch < 4; ++ch) {
            int k0 = ch * 4 + koff;
            v2f a, b;
            a.x = lds[r * RS + base + k0 * 16];
            a.y = lds[r * RS + base + (k0 + 1) * 16];
            b.x = hsign(k0, n);
            b.y = hsign(k0 + 1, n);
            acc = __builtin_amdgcn_wmma_f32_16x16x4_f32(false, a, false, b,
                                                        (short)0, acc, false, false);
        }
        #pragma unroll
        for (int v = 0; v < 8; ++v)
            lds[(v + hi * 8) * RS + base + n * 16] = acc[v];
    }
    __syncthreads();

    // ---- Stage 3: H8 on state bits [10:8]; two columns (m = 2t, 2t+1) packed
    //      per 16x16 tile via block-diagonal B = diag(H8, H8) ----
    #pragma unroll 1
    for (int it = 0; it < 16; ++it) {
        const int m0 = (it * 8 + wave) * 2;
        v8f acc = {};
        #pragma unroll
        for (int ch = 0; ch < 4; ++ch) {
            int k0 = ch * 4 + koff;
            int k1 = k0 + 1;
            v2f a, b;
            a.x = lds[r * RS + (k0 & 7) * 256 + m0 + (k0 >> 3)];
            a.y = lds[r * RS + (k1 & 7) * 256 + m0 + (k1 >> 3)];
            b.x = ((k0 >> 3) == (n >> 3)) ? hsign(k0 & 7, n & 7) : 0.0f;
            b.y = ((k1 >> 3) == (n >> 3)) ? hsign(k1 & 7, n & 7) : 0.0f;
            acc = __builtin_amdgcn_wmma_f32_16x16x4_f32(false, a, false, b,
                                                        (short)0, acc, false, false);
        }
        #pragma unroll
        for (int v = 0; v < 8; ++v)
            lds[(v + hi * 8) * RS + (n & 7) * 256 + m0 + (n >> 3)] = acc[v];
    }
    __syncthreads();

    // ---- |state|^2 with 2^-11 normalization folded in; b128 LDS reads,
    //      float4 coalesced global stores ----
    const float inv = 1.0f / 2048.0f;
    #pragma unroll 1
    for (int it = 0; it < 32; ++it) {
        const int i  = tid + it * 256;
        const int rr = i >> 9;
        const int c4 = i & 511;
        float4 s = *(const float4*)&lds[rr * RS + c4 * 4];
        float4 val;
        val.x = s.x * s.x * inv;
        val.y = s.y * s.y * inv;
        val.z = s.z * s.z * inv;
        val.w = s.w * s.w * inv;
        ((float4*)out)[(size_t)(rowsBase + rr) * 512 + c4] = val;
    }
}

extern "C" void kernel_launch(void* const* d_in, const int* in_sizes, int n_in,
                              void* d_out, int out_size, void* d_ws, size_t ws_size,
                              hipStream_t stream) {
    const float* x1 = (const float*)d_in[0];
    const float* x2 = (const float*)d_in[1];
    float* out = (float*)d_out;

    const int B = in_sizes[0] / 1024;          // 8192 batch rows
    dim3 grid(B / ROWS);                       // 512 workgroups
    dim3 block(256);                           // 8 waves of 32
    size_t ldsBytes = (size_t)ROWS * RS * sizeof(float);  // 131,328 B < 320 KB

    hipLaunchKernelGGL(qwht_fusion_kernel, grid, block, ldsBytes, stream,
                       x1, x2, out);
}